// Conv_57853209477126
// MI455X (gfx1250) — compile-verified
//
#include <hip/hip_runtime.h>

typedef float v2f __attribute__((ext_vector_type(2)));
typedef float v8f __attribute__((ext_vector_type(8)));

// Conv geometry (fixed by the reference)
#define IMG_H 16
#define IMG_W 8
#define OUT_H 14
#define OUT_W 6
#define OUT_ELEMS 84            // 14*6 floats = 336 B (16B aligned per image)
#define WAVES_PER_BLOCK 8
#define GRID_BLOCKS 1024        // 8192 waves -> 32 images per wave (exact)

__device__ __forceinline__ float band_val(const float* kv, int dr, int kk, int n) {
    // M_dr[kk][n] = kernel[dr][kk-n] if 0 <= kk-n <= 2 and n < OUT_W, else 0
    const int d = kk - n;
    float v = 0.0f;
    if (n < OUT_W && d >= 0 && d < 3) v = kv[dr * 3 + d];
    return v;
}

__global__ __launch_bounds__(256) void conv3x3_wmma_f32(
    const float* __restrict__ x,      // [n_imgs, 128]
    const float* __restrict__ kern,   // [9]
    float* __restrict__ out,          // [n_imgs, 84]
    int n_imgs)
{
    __shared__ __align__(16) float lds_img[WAVES_PER_BLOCK][IMG_H * IMG_W];
    __shared__ __align__(16) float lds_stg[WAVES_PER_BLOCK][OUT_ELEMS];

    const int tid  = threadIdx.x;
    const int wv   = tid >> 5;            // wave id in block (wave32)
    const int lane = tid & 31;
    const int m    = lane & 15;           // A row / B column for this lane
    const int hi   = lane >> 4;           // lane-group -> K pair {0,1} vs {2,3}

    const long nwaves = (long)gridDim.x * WAVES_PER_BLOCK;
    long img = (long)blockIdx.x * WAVES_PER_BLOCK + wv;

    // ---- one-time, loop-invariant setup -------------------------------
    float kv[9];
#pragma unroll
    for (int i = 0; i < 9; ++i) kv[i] = kern[i];   // uniform scalar loads

    // B fragments: band matrices for the three kernel rows, two K-chunks
    v2f bf[3][2];
#pragma unroll
    for (int dr = 0; dr < 3; ++dr)
#pragma unroll
        for (int kc = 0; kc < 2; ++kc) {
            const int kk0 = 4 * kc + 2 * hi;
            bf[dr][kc].x = band_val(kv, dr, kk0,     m);
            bf[dr][kc].y = band_val(kv, dr, kk0 + 1, m);
        }

    // A-fragment LDS row pointers (row clamped -> dead D rows, no OOB)
    float* const my_img = lds_img[wv];
    float* const stg    = lds_stg[wv];
    const float* ap[3];
#pragma unroll
    for (int dr = 0; dr < 3; ++dr) {
        const int row = (m + dr < IMG_H) ? (m + dr) : (IMG_H - 1);
        ap[dr] = my_img + row * IMG_W + 2 * hi;
    }

    if (img >= n_imgs) return;            // wave-uniform

    // ---- software-pipelined grid-stride loop --------------------------
    float4 vin = *(const float4*)(x + img * (IMG_H * IMG_W) + lane * 4);

    while (img < n_imgs) {
        const long nxt = img + nwaves;

        asm volatile("" ::: "memory");
        *(float4*)(&my_img[lane * 4]) = vin;            // ds_store_b128
        if (nxt < n_imgs)                               // wave-uniform branch
            vin = *(const float4*)(x + nxt * (IMG_H * IMG_W) + lane * 4);
        asm volatile("" ::: "memory");

        // 6x v_wmma_f32_16x16x4_f32: Y = sum_dr X[dr:dr+14,:] * M_dr
        v8f acc = {};
#pragma unroll
        for (int dr = 0; dr < 3; ++dr) {
#pragma unroll
            for (int kc = 0; kc < 2; ++kc) {
                v2f a;
                a.x = ap[dr][4 * kc];                   // ds_load (fused 2addr_b64)
                a.y = ap[dr][4 * kc + 1];
                acc = __builtin_amdgcn_wmma_f32_16x16x4_f32(
                    false, a, false, bf[dr][kc], (short)0, acc, false, false);
            }
        }

        // ReLU + scatter D (rows 0..13, cols 0..5) to linear staging
        asm volatile("" ::: "memory");
        if (m < OUT_W) {
#pragma unroll
            for (int i = 0; i < 8; ++i) {
                const int mm = 8 * hi + i;              // D row in acc[i]
                if (mm < OUT_H) stg[mm * OUT_W + m] = fmaxf(acc[i], 0.0f);
            }
        }
        asm volatile("" ::: "memory");

        // coalesced store: 84 floats = 21 lanes x float4
        if (lane < OUT_ELEMS / 4) {
            const float4 vo = *(const float4*)(stg + lane * 4);
            *(float4*)(out + img * OUT_ELEMS + lane * 4) = vo;
        }
        img = nxt;
    }
}

extern "C" void kernel_launch(void* const* d_in, const int* in_sizes, int n_in,
                              void* d_out, int out_size, void* d_ws, size_t ws_size,
                              hipStream_t stream) {
    const float* x    = (const float*)d_in[0];   // [4096,64,128] fp32
    const float* kern = (const float*)d_in[1];   // [3,3] fp32
    float* out        = (float*)d_out;           // [4096,64,84] fp32

    const int n_imgs = in_sizes[0] / (IMG_H * IMG_W);   // 262144
    int blocks = GRID_BLOCKS;
    const int max_blocks = (n_imgs + WAVES_PER_BLOCK - 1) / WAVES_PER_BLOCK;
    if (blocks > max_blocks) blocks = max_blocks;

    hipLaunchKernelGGL(conv3x3_wmma_f32, dim3(blocks), dim3(256), 0, stream,
                       x, kern, out, n_imgs);
}